// denoising_block_6124623364167
// MI455X (gfx1250) — compile-verified
//
#include <hip/hip_runtime.h>

typedef __attribute__((ext_vector_type(2))) float v2f;
typedef __attribute__((ext_vector_type(8))) float v8f;

#define B_  32
#define C_  256
#define H_  64
#define W_  64
#define HW_ (H_ * W_)

// Pixels per block: two image rows fused -> halves per-block weight traffic.
#define PX_ 128
// LDS: smed[k>>2][pixel][k&3]  (floats). Plane stride per K-group:
#define KPLANE_ (PX_ * 4)          // 512 floats
#define SMEM_BYTES_ ((C_ / 4) * KPLANE_ * 4)   // 128 KB

__device__ __forceinline__ void s2f(float& a, float& b) {
    float mn = fminf(a, b);
    b = fmaxf(a, b);
    a = mn;
}

// 20-exchange median-of-9 network.
__device__ __forceinline__ float median9(float v0, float v1, float v2,
                                         float v3, float v4, float v5,
                                         float v6, float v7, float v8) {
    s2f(v0, v3); s2f(v1, v4); s2f(v2, v5);
    s2f(v0, v1); s2f(v0, v2);
    s2f(v4, v5); s2f(v3, v5);
    s2f(v1, v2); s2f(v3, v4);
    s2f(v1, v3); s2f(v1, v6);
    s2f(v4, v6); s2f(v2, v6);
    s2f(v2, v3); s2f(v4, v7); s2f(v2, v4); s2f(v3, v7);
    s2f(v4, v8); s2f(v3, v8); s2f(v3, v4);
    return v4;
}

// Unconditional load with address select; zero if tap is outside the image.
__device__ __forceinline__ float ldz(const float* __restrict__ p, int off, bool valid) {
    float v = p[valid ? off : 0];
    return valid ? v : 0.0f;
}

__global__ __launch_bounds__(256)
void fused_median_conv1x1(const float* __restrict__ x,
                          const float* __restrict__ cw,   // [C_out][C_in] row-major
                          const float* __restrict__ cb,   // [C_out]
                          float* __restrict__ out) {
    extern __shared__ float smed[];    // (C_/4) x PX_ x 4 floats = 128 KB

    const int h0 = blockIdx.x * 2;     // this block covers image rows h0, h0+1
    const int b  = blockIdx.y;
    const int t  = threadIdx.x;

    // ---------------- Phase 1: 3x3 median for two rows, all channels --------
    {
        const int w   = t & 63;        // coalesced across lanes
        const int q   = t >> 6;        // 0..3
        const int r0  = q & 1;         // which of the two rows
        const int cp  = q >> 1;        // channel parity
        const int row = h0 + r0;
        const bool hm = (row > 0), hp = (row < H_ - 1);
        const bool wm = (w > 0),   wp = (w < W_ - 1);
        const int  pix = r0 * 64 + w;  // pixel index within block tile
        const size_t rowbase = (size_t)b * C_ * HW_ + (size_t)row * W_ + w;

        for (int c = cp; c < C_; c += 2) {
            const float* p = x + rowbase + (size_t)c * HW_;
            float a0 = ldz(p, -W_ - 1, hm && wm);
            float a1 = ldz(p, -W_,     hm);
            float a2 = ldz(p, -W_ + 1, hm && wp);
            float a3 = ldz(p, -1,      wm);
            float a4 = p[0];
            float a5 = ldz(p,  1,      wp);
            float a6 = ldz(p,  W_ - 1, hp && wm);
            float a7 = ldz(p,  W_,     hp);
            float a8 = ldz(p,  W_ + 1, hp && wp);
            smed[(c >> 2) * KPLANE_ + pix * 4 + (c & 3)] =
                median9(a0, a1, a2, a3, a4, a5, a6, a7, a8);
        }
    }
    __syncthreads();

    // ---------------- Phase 2: WMMA GEMM  out[256 x 128] = W x med ----------
    const int lane = t & 31;
    const int wv   = t >> 5;               // wave id 0..7
    const int o0   = wv * 32;              // wave owns o-tiles [o0,o0+16),[o0+16,o0+32)
    const int l15  = lane & 15;
    const int koff = (lane < 16) ? 0 : 2;  // ISA 16x4 f32 layout: lane halves split K

    // A operand (global, L2-resident): lane -> row o0(+16)+l15, K pair at koff.
    const v2f* a0p = (const v2f*)(cw + (size_t)(o0 +      l15) * C_ + koff);
    const v2f* a1p = (const v2f*)(cw + (size_t)(o0 + 16 + l15) * C_ + koff);
    // B operand (LDS, conflict-free b64): pixel tile j -> pixels j*16..j*16+15.
    const v2f* bp[8];
#pragma unroll
    for (int j = 0; j < 8; ++j)
        bp[j] = (const v2f*)(&smed[(j * 16 + l15) * 4 + koff]);

    v8f acc[2][8];
#pragma unroll
    for (int i = 0; i < 2; ++i)
#pragma unroll
        for (int j = 0; j < 8; ++j)
            acc[i][j] = (v8f){0.f, 0.f, 0.f, 0.f, 0.f, 0.f, 0.f, 0.f};

#pragma unroll 2
    for (int k0 = 0; k0 < C_; k0 += 4) {
        v2f a0 = a0p[k0 >> 1];
        v2f a1 = a1p[k0 >> 1];
        const int bi = (k0 >> 2) * (KPLANE_ / 2);   // v2f stride per K-group plane
        v2f bv[8];
#pragma unroll
        for (int j = 0; j < 8; ++j) bv[j] = bp[j][bi];
#pragma unroll
        for (int j = 0; j < 8; ++j)
            acc[0][j] = __builtin_amdgcn_wmma_f32_16x16x4_f32(false, a0, false, bv[j],
                                                              (short)0, acc[0][j], false, false);
#pragma unroll
        for (int j = 0; j < 8; ++j)
            acc[1][j] = __builtin_amdgcn_wmma_f32_16x16x4_f32(false, a1, false, bv[j],
                                                              (short)0, acc[1][j], false, false);
    }

    // ---------------- Epilogue: out = x + bias + conv  (batched loads) -----
    // C/D 16x16 f32 layout: VGPR r, lanes 0-15 -> M=r, N=lane; lanes 16-31 -> M=r+8.
    const int madd = (lane < 16) ? 0 : 8;

    float cbv[2][8];
#pragma unroll
    for (int i = 0; i < 2; ++i)
#pragma unroll
        for (int r = 0; r < 8; ++r)
            cbv[i][r] = cb[o0 + i * 16 + madd + r];

#pragma unroll
    for (int i = 0; i < 2; ++i) {
#pragma unroll
        for (int j = 0; j < 8; ++j) {
            const int p   = j * 16 + l15;          // pixel within block tile
            const int row = h0 + (p >> 6);
            const int col = p & 63;
            const size_t base = (size_t)b * C_ * HW_ + (size_t)row * W_ + col;
            const int ob = o0 + i * 16 + madd;
            float xv[8];
#pragma unroll
            for (int r = 0; r < 8; ++r)            // one load clause, one wait
                xv[r] = x[base + (size_t)(ob + r) * HW_];
#pragma unroll
            for (int r = 0; r < 8; ++r)
                out[base + (size_t)(ob + r) * HW_] = xv[r] + cbv[i][r] + acc[i][j][r];
        }
    }
}

extern "C" void kernel_launch(void* const* d_in, const int* in_sizes, int n_in,
                              void* d_out, int out_size, void* d_ws, size_t ws_size,
                              hipStream_t stream) {
    (void)in_sizes; (void)n_in; (void)d_ws; (void)ws_size; (void)out_size;
    const float* x  = (const float*)d_in[0];
    const float* cw = (const float*)d_in[1];
    const float* cb = (const float*)d_in[2];
    float* out = (float*)d_out;

    dim3 grid(H_ / 2, B_);   // one block per (2 rows, batch)
    dim3 block(256);         // 8 wave32
    fused_median_conv1x1<<<grid, block, SMEM_BYTES_, stream>>>(x, cw, cb, out);
}